// ImageDecoder_56538949484607
// MI455X (gfx1250) — compile-verified
//
#include <hip/hip_runtime.h>
#include <math.h>

// ---------------------------------------------------------------------------
// ImageDecoder for MI455X (gfx1250).
// ~772MB traffic -> ~33us HBM floor @ 23.3TB/s; VALU trimmed to stay near it.
// One wave32 per image: lane = column (yy), unrolled loop over 32 rows (xx).
// gfx1250 paths: async global->LDS DMA of the noise arrays (overlapped with
// the ALU layers), wave32 readlane/shuffle hoists, per-block shared atan2
// table, non-temporal 256MB output stores.
// ---------------------------------------------------------------------------

#define WAVES_PB 4
#define BLOCK_T  (WAVES_PB * 32)

#if __has_builtin(__builtin_amdgcn_global_load_async_to_lds_b128)
#define USE_ASYNC_LDS 1
#else
#define USE_ASYNC_LDS 0
#endif

typedef int v4i __attribute__((vector_size(16)));

// Builtin params are v4i* in AS1 / AS3. Build the AS pointers by value
// (inttoptr): AS1 = 64-bit global address, AS3 = 32-bit LDS offset
// (low 32 bits of the generic shared-memory address).
#define AS1P(p) ((__attribute__((address_space(1))) v4i*)(unsigned long long)(p))
#define AS3P(p) ((__attribute__((address_space(3))) v4i*)(unsigned int)(unsigned long long)(p))

__global__ __launch_bounds__(BLOCK_T) void
ImageDecoder_56538949484607_kernel(const float* __restrict__ W,
                                   const float* __restrict__ noise_randn,
                                   const float* __restrict__ noise_rand,
                                   float* __restrict__ out,
                                   int nimg)
{
#if USE_ASYNC_LDS
    __shared__ float lds[WAVES_PB * 2048];   // per wave: 1024 randn + 1024 rand (8KB)
#endif
    __shared__ float thetaT[1024];           // image-independent atan2(yy-16, xx-16)

    const int lane = threadIdx.x & 31;
    const int wave = ((int)__builtin_amdgcn_readfirstlane(threadIdx.x)) >> 5;  // wave-uniform
    const int b    = blockIdx.x * WAVES_PB + wave;

    // ---- kick off async DMA of this image's noise into LDS (gfx1250 path) ----
#if USE_ASYNC_LDS
    if (b < nimg) {
        const char* gr = (const char*)(noise_randn + (size_t)b * 1024);
        const char* gu = (const char*)(noise_rand  + (size_t)b * 1024);
        const char* lr = (const char*)&lds[wave * 2048];
        const char* lu = (const char*)&lds[wave * 2048 + 1024];
#pragma unroll
        for (int t = 0; t < 8; ++t) {
            const int off = (t * 32 + lane) * 16;   // 512B per wave per issue
            __builtin_amdgcn_global_load_async_to_lds_b128(AS1P(gr + off), AS3P(lr + off), 0, 0);
            __builtin_amdgcn_global_load_async_to_lds_b128(AS1P(gu + off), AS3P(lu + off), 0, 0);
        }
    }
#endif

    // ---- per-block geometry table: theta = atan2(yy-CY, xx-CX), 8 per thread ----
#pragma unroll
    for (int t = 0; t < 1024 / BLOCK_T; ++t) {
        const int p = t * BLOCK_T + threadIdx.x;
        thetaT[p] = atan2f((float)(p & 31) - 16.0f, (float)(p >> 5) - 16.0f);
    }
    __syncthreads();
    if (b >= nimg) return;

    // ---- per-image latents (wave-uniform -> scalar loads) ----
    const float* Wp = W + (size_t)b * 16;
    const float c0 = Wp[0],  c1 = Wp[1],  c2 = Wp[2],  c3 = Wp[3];
    const float c4 = Wp[4],  c5 = Wp[5],  c6 = Wp[6],  c7 = Wp[7];
    const float c8 = Wp[8],  c9 = Wp[9],  c10 = Wp[10], c11 = Wp[11];
    const float c12 = Wp[12], c13 = Wp[13], c14 = Wp[14], c15 = Wp[15];

    const float PI = 3.14159265358979323846f;

    // per-image precompute
    const float half_  = floorf(fminf(fmaxf(fabsf(c1) * 8.0f, 2.0f), 12.0f));
    const float sq_lo  = 16.0f - half_, sq_hi = 16.0f + half_;
    const float bx     = truncf(16.0f + c7 * 5.0f);          // by == bx
    const float sigma  = 4.0f + fabsf(c7) * 4.0f;
    const float negInv2s2 = -1.0f / (2.0f * sigma * sigma);
    const float cbf    = floorf(fminf(fmaxf(fabsf(c9) * 8.0f + 2.0f, 2.0f), 16.0f));
    const int   cbi    = (int)cbf;
    const float angle  = c10 * PI;
    const float ca     = __cosf(angle), sa = __sinf(angle);
    const float ring_c = c8 * 10.0f;
    const float k2     = c2 * 10.0f * PI / 32.0f;
    const float k3     = c3 * 10.0f * PI / 32.0f;
    const float k4     = c4 * 5.0f * PI / 64.0f;
    const float k11    = c11 * 5.0f;
    const float cone_a = c0 * 5.0f;

    // per-lane (yy = lane) precompute; lane index domain == row index domain,
    // so row-side factors are obtained below with v_readlane (__shfl const idx).
    const float fl   = (float)lane;          // yy
    const float dyc  = fl - 16.0f;           // yy - CY
    const float dyc2 = dyc * dyc;
    const float sin2L = 0.5f * __sinf(k2 * fl);            // L2 row term (via shfl)
    const float sin3  = 0.5f * __sinf(k3 * fl);            // L3 lane term
    const float sin5  = 0.5f * __sinf((fl + c5 * 10.0f) * (PI / 32.0f));   // L5 lane term
    const float laneBias = sin3 + sin5 + c6;               // L3 + L5 + L6 folded
    float gq = fl - bx;
    const float gL = __expf(gq * gq * negInv2s2);          // gaussian lane/row factor
    const float s4L = __sinf(k4 * fl), c4L = __cosf(k4 * fl); // L4 identity factors
    const float yHit = (fl >= sq_lo && fl < sq_hi) ? 0.5f : 0.0f;
    const int   qy   = lane / cbi;                         // exact checker quotient
    const float sa_dy = sa * dyc;
    const float* thL = &thetaT[lane];

    float val[32];

    // ---- layers 0..11 (overlaps the async DMA above) ----
#pragma unroll
    for (int i = 0; i < 32; ++i) {
        const float fx  = (float)i;            // xx (compile-time literal)
        const float dxc = fx - 16.0f;
        const float r   = sqrtf(dxc * dxc + dyc2);          // dxc*dxc is a literal

        float v = fminf(fmaxf(cone_a - r, 0.0f), 1.0f);     // L0 cone
        v += (fx >= sq_lo && fx < sq_hi) ? yHit : 0.0f;     // L1 square
        v += __shfl(sin2L, i, 32);                          // L2 (row sinusoid)
        v += laneBias;                                      // L3 + L5 + L6
        const float s4i = __shfl(s4L, i, 32);               // L4: sin(k4(i+l)) =
        const float c4i = __shfl(c4L, i, 32);               //   s4i*c4l + c4i*s4l
        v += 0.5f * (s4i * c4L + c4i * s4L);
        v += __shfl(gL, i, 32) * gL;                        // L7 gaussian (separated)
        const float rr = r - ring_c;
        v += (rr * rr < 10.0f) ? 1.0f : 0.0f;               // L8 ring
        const int qx = __shfl(qy, i, 32);                   // floor(xx/cb) exactly
        v += ((qx + qy) & 1) ? 0.3f : 0.0f;                 // L9 checker
        const float rot = ca * dxc + sa_dy;                 // L10 line
        v += (fabsf(rot) < 3.0f) ? 0.6f : 0.0f;
        v += 0.5f * __sinf(thL[i * 32] * k11);              // L11 angular (LDS theta)
        val[i] = v;
    }

    // ---- noise now needed: complete the async DMA ----
#if USE_ASYNC_LDS
#if __has_builtin(__builtin_amdgcn_s_wait_asynccnt)
    __builtin_amdgcn_s_wait_asynccnt(0);
#else
    asm volatile("s_wait_asynccnt 0x0" ::: "memory");
#endif
    const float* nr = &lds[wave * 2048];
    const float* nu = &lds[wave * 2048 + 1024];
#else
    const float* nr = noise_randn + (size_t)b * 1024;
    const float* nu = noise_rand  + (size_t)b * 1024;
#endif

    // ---- layers 12..15 + per-image min/max ----
    const float a12 = 0.1f * c12;
    const float a13 = 0.05f * c13;
    const float con = 1.0f + c14;
    const bool  inv = (c15 > 0.0f);
    float vmin = 1e30f, vmax = -1e30f;
#pragma unroll
    for (int i = 0; i < 32; ++i) {
#if USE_ASYNC_LDS
        const float rn = nr[i * 32 + lane];
        const float ru = nu[i * 32 + lane];
#else
        const float rn = __builtin_nontemporal_load(nr + i * 32 + lane);
        const float ru = __builtin_nontemporal_load(nu + i * 32 + lane);
#endif
        float v = val[i] + rn * a12 + (ru - 0.5f) * a13;
        v = (v - 0.5f) * con + 0.5f;
        if (inv) v = 1.0f - v;
        val[i] = v;
        vmin = fminf(vmin, v);
        vmax = fmaxf(vmax, v);
    }

    // ---- wave32 min/max reduction (no LDS, no barrier) ----
#pragma unroll
    for (int o = 16; o > 0; o >>= 1) {
        vmin = fminf(vmin, __shfl_xor(vmin, o, 32));
        vmax = fmaxf(vmax, __shfl_xor(vmax, o, 32));
    }
    const float scale = 1.0f / (vmax - vmin + 1e-8f);

    // ---- normalize + streaming (non-temporal) store ----
    float* op = out + (size_t)b * 1024 + lane;
#pragma unroll
    for (int i = 0; i < 32; ++i) {
        __builtin_nontemporal_store((val[i] - vmin) * scale, op + i * 32);
    }
}

extern "C" void kernel_launch(void* const* d_in, const int* in_sizes, int n_in,
                              void* d_out, int out_size, void* d_ws, size_t ws_size,
                              hipStream_t stream) {
    (void)n_in; (void)out_size; (void)d_ws; (void)ws_size;
    const float* W  = (const float*)d_in[0];
    const float* nr = (const float*)d_in[1];
    const float* nu = (const float*)d_in[2];
    float* out = (float*)d_out;
    const int B = in_sizes[0] / 16;
    const int blocks = (B + WAVES_PB - 1) / WAVES_PB;
    ImageDecoder_56538949484607_kernel<<<blocks, BLOCK_T, 0, stream>>>(W, nr, nu, out, B);
}